// GraphNeuralNetworkBlockSelect_38517266710691
// MI455X (gfx1250) — compile-verified
//
#include <hip/hip_runtime.h>
#include <hip/hip_bf16.h>

// ---------------------------------------------------------------------------
// GNN block: GAT-style masked edge softmax + aggregation + Linear(2H->H)+ReLU
// N=20000 nodes, E=320000 edges, T=4, H=128.  All fp32 (reference dtype).
// Final GEMM uses V_WMMA_F32_16X16X4_F32 (CDNA5 wave32 WMMA).
// ---------------------------------------------------------------------------

#define N_NODES 20000
#define N_EDGES 320000
#define TT      4
#define HH      128
#define NT      (N_NODES * TT)        // 80000 rows
#define ET      (N_EDGES * TT)        // 1280000 edge-timesteps
#define OUT_ELEMS (NT * HH)           // 10,240,000

typedef __attribute__((ext_vector_type(2))) float v2f;
typedef __attribute__((ext_vector_type(8))) float v8f;

// ---- float atomic max via int ordering trick (native i32/u32 L2 atomics) ---
__device__ __forceinline__ void atomicMaxFloat(float* addr, float val) {
    if (val >= 0.0f)
        atomicMax((int*)addr, __float_as_int(val));
    else
        atomicMin((unsigned int*)addr, (unsigned int)__float_as_int(val));
}

// ---------------------------------------------------------------------------
// 0) init: zero d_out (agg accumulator), m = -1e9, den = 0
// ---------------------------------------------------------------------------
__global__ __launch_bounds__(256) void k_init(float* __restrict__ out,
                                              float* __restrict__ m,
                                              float* __restrict__ den) {
    int i = blockIdx.x * 256 + threadIdx.x;
    if (i < OUT_ELEMS) out[i] = 0.0f;
    if (i < NT) { m[i] = -1e9f; den[i] = 0.0f; }
}

// ---------------------------------------------------------------------------
// 1) per-(node,timestep) attention projections: one wave per row of nf
//    proj_s[r] = nf[r,:] . a_src ; proj_d[r] = nf[r,:] . a_dst
// ---------------------------------------------------------------------------
__global__ __launch_bounds__(256) void k_proj(const float* __restrict__ nf,
                                              const float* __restrict__ a_src,
                                              const float* __restrict__ a_dst,
                                              float* __restrict__ proj_s,
                                              float* __restrict__ proj_d) {
    int wave = blockIdx.x * 8 + (threadIdx.x >> 5);   // row index in [0, NT)
    int lane = threadIdx.x & 31;
    if (wave >= NT) return;
    const float4* nfv = (const float4*)(nf + (size_t)wave * HH);
    const float4* asv = (const float4*)a_src;
    const float4* adv = (const float4*)a_dst;
    float4 x = nfv[lane];
    float4 as = asv[lane];
    float4 ad = adv[lane];
    float ss = x.x * as.x + x.y * as.y + x.z * as.z + x.w * as.w;
    float sd = x.x * ad.x + x.y * ad.y + x.z * ad.z + x.w * ad.w;
    #pragma unroll
    for (int off = 16; off > 0; off >>= 1) {
        ss += __shfl_xor(ss, off, 32);
        sd += __shfl_xor(sd, off, 32);
    }
    if (lane == 0) { proj_s[wave] = ss; proj_d[wave] = sd; }
}

// ---------------------------------------------------------------------------
// 2) edge pass 1: masked leaky-relu logits; segment max into m[dst,t]
//    thread per (edge, t); logit stored in exbuf for pass 2
// ---------------------------------------------------------------------------
__global__ __launch_bounds__(256) void k_edge_logit_max(const int* __restrict__ src,
                                                        const int* __restrict__ dst,
                                                        const int* __restrict__ events,
                                                        const float* __restrict__ proj_s,
                                                        const float* __restrict__ proj_d,
                                                        float* __restrict__ exbuf,
                                                        float* __restrict__ m) {
    int i = blockIdx.x * 256 + threadIdx.x;
    if (i >= ET) return;
    int e = i >> 2, t = i & 3;
    int s = src[e], d = dst[e];
    float x = proj_s[s * TT + t] + proj_d[d * TT + t];
    float lg = (x > 0.0f) ? x : 0.2f * x;           // leaky_relu(0.2)
    bool active = (events[s * TT + t] == 1);
    lg = active ? lg : -1e9f;
    exbuf[i] = lg;
    atomicMaxFloat(&m[d * TT + t], lg);
}

// ---------------------------------------------------------------------------
// 3) edge pass 2: ex = active ? exp(logit - m[dst]) : 0 ; segment sum -> den
// ---------------------------------------------------------------------------
__global__ __launch_bounds__(256) void k_edge_exp_sum(const int* __restrict__ src,
                                                      const int* __restrict__ dst,
                                                      const int* __restrict__ events,
                                                      const float* __restrict__ m,
                                                      float* __restrict__ exbuf,
                                                      float* __restrict__ den) {
    int i = blockIdx.x * 256 + threadIdx.x;
    if (i >= ET) return;
    int e = i >> 2, t = i & 3;
    int s = src[e], d = dst[e];
    bool active = (events[s * TT + t] == 1);
    float ex = 0.0f;
    if (active) ex = expf(exbuf[i] - m[d * TT + t]);
    exbuf[i] = ex;
    if (ex != 0.0f) atomicAdd(&den[d * TT + t], ex);
}

// ---------------------------------------------------------------------------
// 4) edge pass 3: agg[dst,t,:] += alpha * nf[src,t,:]  (one wave per (e,t))
//    agg lives directly in d_out.
// ---------------------------------------------------------------------------
__global__ __launch_bounds__(256) void k_edge_aggregate(const int* __restrict__ src,
                                                        const int* __restrict__ dst,
                                                        const float* __restrict__ exbuf,
                                                        const float* __restrict__ den,
                                                        const float* __restrict__ nf,
                                                        float* __restrict__ out) {
    int wave = blockIdx.x * 8 + (threadIdx.x >> 5);   // (e,t) index
    int lane = threadIdx.x & 31;
    if (wave >= ET) return;
    int e = wave >> 2, t = wave & 3;
    float ex = exbuf[wave];
    if (ex == 0.0f) return;                           // uniform across wave
    int d = dst[e];
    float alpha = ex / fmaxf(den[d * TT + t], 1e-9f);
    int s = src[e];
    const float4* nfv = (const float4*)(nf + ((size_t)s * TT + t) * HH);
    float4 v = nfv[lane];
    float* dstp = out + ((size_t)d * TT + t) * HH + lane * 4;
    atomicAdd(dstp + 0, alpha * v.x);
    atomicAdd(dstp + 1, alpha * v.y);
    atomicAdd(dstp + 2, alpha * v.z);
    atomicAdd(dstp + 3, alpha * v.w);
}

// ---------------------------------------------------------------------------
// 5) node update: out = relu([nf | agg] @ W + b), in place over d_out (agg).
//    Block = 256 threads = 8 waves, handles 16 rows x 128 cols.
//    Wave w computes one 16x16 tile with V_WMMA_F32_16X16X4_F32, K-loop of 64.
//    A (16x4 f32): lane l -> M = l&15, K = kb*4 + (l>>4)*2 + {0,1} (2 VGPRs)
//    B (4x16 f32): lane l -> N = l&15, K = kb*4 + (l>>4)*2 + {0,1} (2 VGPRs)
//    C (16x16 f32): vgpr j, lane l -> M = j + (l>>4)*8, N = l&15
// ---------------------------------------------------------------------------
#define LDS_PITCH 257   // 257 % 64 == 1 -> conflict-free column reads

__global__ __launch_bounds__(256) void k_gemm_relu(const float* __restrict__ nf,
                                                   const float* __restrict__ W,
                                                   const float* __restrict__ bias,
                                                   float* __restrict__ out) {
    __shared__ float tile[16 * LDS_PITCH];
    const int rb  = blockIdx.x;           // 16-row slab
    const int tid = threadIdx.x;

    // cooperative stage of X = [nf rows (k<128) | agg rows (k>=128, from out)]
    for (int idx = tid; idx < 16 * 256; idx += 256) {
        int r = idx >> 8, k = idx & 255;
        size_t row = (size_t)rb * 16 + r;
        float v = (k < HH) ? nf[row * HH + k] : out[row * HH + (k - HH)];
        tile[r * LDS_PITCH + k] = v;
    }
    __syncthreads();

    const int w    = tid >> 5;
    const int lane = tid & 31;
    const int mrow = lane & 15;
    const int koff = (lane >> 4) * 2;
    const int ncol = w * 16 + (lane & 15);

    v8f c = {};
    #pragma unroll 4
    for (int kb = 0; kb < 64; ++kb) {
        int k0 = kb * 4 + koff;
        v2f a, bvec;
        a.x    = tile[mrow * LDS_PITCH + k0];
        a.y    = tile[mrow * LDS_PITCH + k0 + 1];
        bvec.x = W[(size_t)k0 * HH + ncol];
        bvec.y = W[(size_t)(k0 + 1) * HH + ncol];
        c = __builtin_amdgcn_wmma_f32_16x16x4_f32(
                /*neg_a=*/false, a, /*neg_b=*/false, bvec,
                /*c_mod=*/(short)0, c, /*reuse_a=*/false, /*reuse_b=*/false);
    }

    float bv = bias[ncol];
    size_t mbase = (size_t)rb * 16 + ((lane >> 4) * 8);
    #pragma unroll
    for (int j = 0; j < 8; ++j) {
        float v = c[j] + bv;
        out[(mbase + j) * HH + ncol] = v > 0.0f ? v : 0.0f;
    }
}

// ---------------------------------------------------------------------------
// launcher
// ---------------------------------------------------------------------------
extern "C" void kernel_launch(void* const* d_in, const int* in_sizes, int n_in,
                              void* d_out, int out_size, void* d_ws, size_t ws_size,
                              hipStream_t stream) {
    const float* nf     = (const float*)d_in[0];   // [N,T,H]
    const int*   events = (const int*)  d_in[1];   // [N,T]
    const int*   src    = (const int*)  d_in[2];   // [E]
    const int*   dst    = (const int*)  d_in[3];   // [E]
    const float* a_src  = (const float*)d_in[4];   // [H]
    const float* a_dst  = (const float*)d_in[5];   // [H]
    const float* W      = (const float*)d_in[6];   // [2H,H]
    const float* b      = (const float*)d_in[7];   // [H]
    float* out = (float*)d_out;                    // [N,T,H]; also agg scratch

    // workspace layout (floats): proj_s | proj_d | m | den | exbuf
    float* proj_s = (float*)d_ws;
    float* proj_d = proj_s + NT;
    float* m      = proj_d + NT;
    float* den    = m + NT;
    float* exbuf  = den + NT;                      // ET floats; total ~6.4 MB

    k_init<<<OUT_ELEMS / 256, 256, 0, stream>>>(out, m, den);
    k_proj<<<NT / 8, 256, 0, stream>>>(nf, a_src, a_dst, proj_s, proj_d);
    k_edge_logit_max<<<ET / 256, 256, 0, stream>>>(src, dst, events,
                                                   proj_s, proj_d, exbuf, m);
    k_edge_exp_sum<<<ET / 256, 256, 0, stream>>>(src, dst, events, m, exbuf, den);
    k_edge_aggregate<<<ET / 8, 256, 0, stream>>>(src, dst, exbuf, den, nf, out);
    k_gemm_relu<<<NT / 16, 256, 0, stream>>>(nf, W, b, out);
}